// RHS_layer_34102040330828
// MI455X (gfx1250) — compile-verified
//
#include <hip/hip_runtime.h>
#include <hip/hip_bf16.h>

typedef __attribute__((ext_vector_type(2))) float v2f;
typedef __attribute__((ext_vector_type(8))) float v8f;

// S^T padded to [12][16], k-major: c_ST[k*16 + n] = S[n][k] (n < 5), 0 otherwise.
__device__ __constant__ float c_ST[12 * 16] = {
    -2.f,  1.f,  0.f,  0.f,  0.f, 0,0,0,0,0,0,0,0,0,0,0,   // k = 0
     2.f, -1.f,  0.f,  0.f,  0.f, 0,0,0,0,0,0,0,0,0,0,0,   // k = 1
    -1.f,  0.f,  1.f, -1.f,  0.f, 0,0,0,0,0,0,0,0,0,0,0,   // k = 2
     1.f,  0.f, -1.f,  1.f,  0.f, 0,0,0,0,0,0,0,0,0,0,0,   // k = 3
     0.f,  0.f,  0.f, -2.f,  1.f, 0,0,0,0,0,0,0,0,0,0,0,   // k = 4
     0.f,  0.f,  0.f,  2.f, -1.f, 0,0,0,0,0,0,0,0,0,0,0,   // k = 5
     0.f, -1.f,  2.f,  0.f, -1.f, 0,0,0,0,0,0,0,0,0,0,0,   // k = 6
     0.f,  1.f, -2.f,  0.f,  1.f, 0,0,0,0,0,0,0,0,0,0,0,   // k = 7
     1.f, -1.f,  1.f, -1.f,  0.f, 0,0,0,0,0,0,0,0,0,0,0,   // k = 8
    -1.f,  1.f, -1.f,  1.f,  0.f, 0,0,0,0,0,0,0,0,0,0,0,   // k = 9
    -1.f,  0.f,  1.f,  1.f, -1.f, 0,0,0,0,0,0,0,0,0,0,0,   // k = 10
     1.f,  0.f, -1.f, -1.f,  1.f, 0,0,0,0,0,0,0,0,0,0,0    // k = 11
};

#define LOG2_10 3.3219280948873623f

// One wave handles one 16-row tile per loop iteration.
// A (16x4 f32, WMMA j): lane l<16 holds row l with K = 4j+{0,1};
//                       lane l+16 holds row l with K = 4j+{2,3}.
// B (4x16 f32), mirrored layout: vgpr v, lane -> K = 4j + 2*(lane>=16) + v,
//   N = lane&15. B carries 10^w[k] * S[n][k]; built once outside the loop.
// D (16x16 f32): vgpr r, lanes 0-15 = (M=r, N=lane), lanes 16-31 = (M=8+r).
__global__ __launch_bounds__(256)
void rhs_wmma_kernel(const float* __restrict__ x, const float* __restrict__ w,
                     float* __restrict__ out, int numTiles) {
    const int lane = threadIdx.x & 31;
    const bool h   = (lane & 16) != 0;   // half-wave
    const int n    = lane & 15;          // N column / row-within-tile

    const int wavesPerBlock = blockDim.x >> 5;
    const int globalWave = blockIdx.x * wavesPerBlock + (threadIdx.x >> 5);
    const int totalWaves = gridDim.x * wavesPerBlock;

    // ---- Build B operands once: B[j][v] = 10^w[k] * S^T[k][n], k = 4j+2*h+v
    v2f Bm[3];
#pragma unroll
    for (int j = 0; j < 3; ++j) {
#pragma unroll
        for (int v = 0; v < 2; ++v) {
            const int k = 4 * j + (h ? 2 : 0) + v;
            const float s10 = exp2f(w[k] * LOG2_10);
            Bm[j][v] = c_ST[k * 16 + n] * s10;
        }
    }

    for (int t = globalWave; t < numTiles; t += totalWaves) {
        const int rowBase = t * 16;
        const float* xp = x + (size_t)(rowBase + n) * 5;  // both halves: same 16 rows
        const float z0 = xp[0], z1 = xp[1], z2 = xp[2], z3 = xp[3], z4 = xp[4];

        // A operands, explicit selects on half (no arrays -> no extract trees):
        //   half0 needs k = {0,1, 4,5, 8,9} = {z0*z0, z1, z3*z3, z4, z1*z3, z2*z0}
        //   half1 needs k = {2,3, 6,7,10,11} = {z0*z3, z2, z1*z4, z2*z2, z4*z0, z2*z3}
        const float u   = h ? z3 : z0;          // shared by a00 and a21
        const float a00 = z0 * u;               // z0*z0  | z0*z3
        const float a01 = h ? z2 : z1;          // z1     | z2
        const float s1a = h ? z1 : z3;
        const float s1b = h ? z4 : z3;
        const float a10 = s1a * s1b;            // z3*z3  | z1*z4
        const float z22 = z2 * z2;
        const float a11 = h ? z22 : z4;         // z4     | z2*z2
        const float s2a = h ? z4 : z1;
        const float s2b = h ? z0 : z3;
        const float a20 = s2a * s2b;            // z1*z3  | z4*z0
        const float a21 = z2 * u;               // z2*z0  | z2*z3

        v8f d = {0.f, 0.f, 0.f, 0.f, 0.f, 0.f, 0.f, 0.f};
        v2f a;
        a[0] = a00; a[1] = a01;
        d = __builtin_amdgcn_wmma_f32_16x16x4_f32(false, a, false, Bm[0],
                                                  (short)0, d, false, false);
        a[0] = a10; a[1] = a11;
        d = __builtin_amdgcn_wmma_f32_16x16x4_f32(false, a, false, Bm[1],
                                                  (short)0, d, false, false);
        a[0] = a20; a[1] = a21;
        d = __builtin_amdgcn_wmma_f32_16x16x4_f32(false, a, false, Bm[2],
                                                  (short)0, d, false, false);

        // Scatter D: vgpr r -> out row (rowBase + r + 8*half), col n, n<5 only
        if (n < 5) {
            float* op = out + ((size_t)(rowBase + (h ? 8 : 0))) * 5 + n;
#pragma unroll
            for (int r = 0; r < 8; ++r)
                op[(size_t)r * 5] = d[r];
        }
    }
}

// Scalar fallback for a non-multiple-of-16 tail (never launched for BATCH=4M).
__global__ __launch_bounds__(256)
void rhs_tail_kernel(const float* __restrict__ x, const float* __restrict__ w,
                     float* __restrict__ out, int startRow, int nRows) {
    const int i = blockIdx.x * blockDim.x + threadIdx.x;
    if (i >= nRows) return;
    const int row = startRow + i;
    const float* xp = x + (size_t)row * 5;
    const float z0 = xp[0], z1 = xp[1], z2 = xp[2], z3 = xp[3], z4 = xp[4];
    float p[12];
    p[0] = z0 * z0;  p[1]  = z1;       p[2]  = z0 * z3;  p[3]  = z2;
    p[4] = z3 * z3;  p[5]  = z4;       p[6]  = z1 * z4;  p[7]  = z2 * z2;
    p[8] = z1 * z3;  p[9]  = z2 * z0;  p[10] = z4 * z0;  p[11] = z2 * z3;
    float acc[5] = {0.f, 0.f, 0.f, 0.f, 0.f};
#pragma unroll
    for (int k = 0; k < 12; ++k) {
        const float f = p[k] * exp2f(w[k] * LOG2_10);
#pragma unroll
        for (int nn = 0; nn < 5; ++nn) acc[nn] += f * c_ST[k * 16 + nn];
    }
#pragma unroll
    for (int nn = 0; nn < 5; ++nn) out[(size_t)row * 5 + nn] = acc[nn];
}

extern "C" void kernel_launch(void* const* d_in, const int* in_sizes, int n_in,
                              void* d_out, int out_size, void* d_ws, size_t ws_size,
                              hipStream_t stream) {
    const float* x = (const float*)d_in[0];   // [B,5] fp32
    const float* w = (const float*)d_in[1];   // [12]  fp32
    float* out = (float*)d_out;               // [B,5] fp32

    const long batch = (long)in_sizes[0] / 5;
    const int numTiles = (int)(batch / 16);
    const int rem = (int)(batch - (long)numTiles * 16);

    if (numTiles > 0) {
        const int threads = 256;              // 8 waves/block
        int blocks = 2048;                    // grid-stride over tiles
        const int maxBlocks = (numTiles + 7) / 8;
        if (blocks > maxBlocks) blocks = maxBlocks;
        rhs_wmma_kernel<<<blocks, threads, 0, stream>>>(x, w, out, numTiles);
    }
    if (rem > 0) {
        rhs_tail_kernel<<<(rem + 255) / 256, 256, 0, stream>>>(x, w, out,
                                                               numTiles * 16, rem);
    }
}